// FrequencyDecomposition_61770219651206
// MI455X (gfx1250) — compile-verified
//
#include <hip/hip_runtime.h>
#include <hip/hip_bf16.h>

// fp32 WMMA operand/accumulator vectors (wave32: 16x4 A = 2 VGPRs, 16x16 C/D = 8 VGPRs)
typedef __attribute__((ext_vector_type(2))) float v2f;
typedef __attribute__((ext_vector_type(4))) float v4f;
typedef __attribute__((ext_vector_type(8))) float v8f;

#define LDSS  20         // padded LDS row stride in dwords (bank-conflict free for our patterns)
#define WPB   8          // waves per 256-thread block
#define PLANE 262144     // 512*512
#define NTOT  12582912   // 16*3*512*512 (one output tensor)

// Standard 8x8 JPEG zig-zag traversal index (matches reference _zigzag_indices(8))
__device__ __constant__ int ZZ8[64] = {
   0,  1,  5,  6, 14, 15, 27, 28,
   2,  4,  7, 13, 16, 26, 29, 42,
   3,  8, 12, 17, 25, 30, 41, 43,
   9, 11, 18, 24, 31, 40, 44, 53,
  10, 19, 23, 32, 39, 45, 52, 54,
  20, 22, 33, 38, 46, 51, 55, 60,
  21, 34, 37, 47, 50, 56, 59, 61,
  35, 36, 48, 49, 57, 58, 62, 63
};

// 8x8 DCT-II matrix element: D[k][i]
__device__ __forceinline__ float dctD(int k, int i) {
  if (k == 0) return 0.35355339059327373f;                         // 1/sqrt(8)
  return 0.5f * cospif((float)(k * (2 * i + 1)) * (1.0f / 16.0f)); // sqrt(2/8)*cos(pi*k*(2i+1)/16)
}
// Block-diagonal 16x16: D2 = diag(D, D)
__device__ __forceinline__ float d2e(int r, int c) {
  if ((r ^ c) & 8) return 0.0f;
  return dctD(r & 7, c & 7);
}

__device__ __forceinline__ v8f wmma4(v2f a, v2f b, v8f c) {
  // D = A(16x4,f32) x B(4x16,f32) + C(16x16,f32)
  return __builtin_amdgcn_wmma_f32_16x16x4_f32(false, a, false, b, (short)0, c, false, false);
}

// ---------------------------------------------------------------------------
// Init kernel (1 wave, runs once per launch): bake per-lane operand tables.
//   tabPQ[kc*32+lane] = { cP.x, cP.y, cQ.x, cQ.y }
//   tabM [kc*32+lane] = { mLow.x, mLow.y, mHigh.x, mHigh.y }
// ---------------------------------------------------------------------------
extern "C" __global__ void freq_tabs_init(v4f* __restrict__ ws)
{
  const int lane = threadIdx.x & 31;
  const int col = lane & 15;
  const int h2  = (lane >> 4) << 1;
  const int c7  = lane & 7;
  v4f* tabPQ = ws;
  v4f* tabM  = ws + 4 * 32;
#pragma unroll
  for (int kc = 0; kc < 4; ++kc) {
    const int k0 = 4 * kc + h2;
    v4f pq, m;
    pq.x = d2e(col, k0);     pq.y = d2e(col, k0 + 1);   // cP: A-layout(D2) == B-layout(D2^T)
    pq.z = d2e(k0, col);     pq.w = d2e(k0 + 1, col);   // cQ: A-layout(D2^T) == B-layout(D2)
    const int z0 = ZZ8[(k0 & 7) * 8 + c7];
    const int z1 = ZZ8[((k0 + 1) & 7) * 8 + c7];
    m.x = (z0 < 16) ? 1.0f : 0.0f;                      // low mask
    m.y = (z1 < 16) ? 1.0f : 0.0f;
    m.z = (z0 >= 48) ? 1.0f : 0.0f;                     // high mask
    m.w = (z1 >= 48) ? 1.0f : 0.0f;
    tabPQ[kc * 32 + lane] = pq;
    tabM [kc * 32 + lane] = m;
  }
}

// ---------------------------------------------------------------------------
// Main kernel: one 16x16 image tile (= 2x2 grid of 8x8 DCT blocks) per wave.
//   low  = D2^T ((D2 X D2^T) o maskL) D2
//   high = D2^T ((D2 X D2^T) o maskH) D2
//   mid  = x - low - high        (masks partition unity; exact by linearity)
// ---------------------------------------------------------------------------
extern "C" __global__ void __launch_bounds__(256)
freq_decomp_wmma(const float* __restrict__ x, float* __restrict__ out,
                 const v4f* __restrict__ ws)
{
  __shared__ float lds[WPB][16 * LDSS];

  const int lane = threadIdx.x & 31;
  const int wave = threadIdx.x >> 5;
  const int tile = blockIdx.x * WPB + wave;

  const int bc  = tile >> 10;                    // 1024 tiles per 512x512 plane
  const int rem = tile & 1023;
  const int idx = bc * PLANE + (rem >> 5) * 8192 + (rem & 31) * 16;  // tile base element (<12.6M)

  const int col = lane & 15;
  const int h2  = (lane >> 4) << 1;              // 0 or 2  (K offset for upper lane half)
  const int hm  = (lane >> 4) << 3;              // 0 or 8  (M offset in C/D layout)

  // 32-bit per-lane element indices; all global accesses = SGPR base + this + imm
  const int rowIdx = idx + hm * 512 + col;       // C-layout: + v*512
  const int bIdx   = idx + h2 * 512 + col;       // B-layout: + kc*2048 (+512 for .y)

  // Precomputed LDS dword indices; all ds ops = base reg + imm offset
  const int lStore = hm * LDSS + col;            // + v*LDSS
  const int lA     = col * LDSS + h2;            // + kc*4      (b64 pairs)
  const int lB     = h2 * LDSS + col;            // + kc*4*LDSS (2addr pairs, +LDSS)

  // Baked operand tables: 8 coalesced b128 loads (4 KB, L2-resident)
  v2f cP[4], cQ[4], mL[4], mH[4];
  const v4f* tabPQ = ws;
  const v4f* tabM  = ws + 4 * 32;
#pragma unroll
  for (int kc = 0; kc < 4; ++kc) {
    const v4f pq = tabPQ[kc * 32 + lane];
    const v4f m  = tabM [kc * 32 + lane];
    cP[kc].x = pq.x; cP[kc].y = pq.y;
    cQ[kc].x = pq.z; cQ[kc].y = pq.w;
    mL[kc].x = m.x;  mL[kc].y = m.y;
    mH[kc].x = m.z;  mH[kc].y = m.w;
  }

  float* buf = lds[wave];                        // wave-private staging (in-order DS, no barriers)

  // X tile in B-layout (stage-1 operand) and C-layout (mid accumulator + passthrough)
  v2f b1[4];
  float cX[8];
#pragma unroll
  for (int kc = 0; kc < 4; ++kc) {
    b1[kc].x = x[bIdx + kc * 2048];
    b1[kc].y = x[bIdx + kc * 2048 + 512];
  }
#pragma unroll
  for (int v = 0; v < 8; ++v) cX[v] = x[rowIdx + v * 512];

  // x passthrough (output #3) straight from registers
#pragma unroll
  for (int v = 0; v < 8; ++v)
    __builtin_nontemporal_store(cX[v], out + 3 * NTOT + rowIdx + v * 512);

  // Stage 1: T1 = D2 @ X
  v8f t1 = {};
#pragma unroll
  for (int kc = 0; kc < 4; ++kc) t1 = wmma4(cP[kc], b1[kc], t1);

  // Convert T1: C-layout -> A-layout via LDS
#pragma unroll
  for (int v = 0; v < 8; ++v) buf[lStore + v * LDSS] = t1[v];
  v2f aT[4];
#pragma unroll
  for (int kc = 0; kc < 4; ++kc) {
    aT[kc].x = buf[lA + kc * 4];                 // contiguous pair -> ds_load_b64
    aT[kc].y = buf[lA + kc * 4 + 1];
  }

  // Stage 2: DCT = T1 @ D2^T
  v8f dct = {};
#pragma unroll
  for (int kc = 0; kc < 4; ++kc) dct = wmma4(aT[kc], cP[kc], dct);

  // Convert DCT: C-layout -> B-layout via LDS (once; masks applied per band in registers)
#pragma unroll
  for (int v = 0; v < 8; ++v) buf[lStore + v * LDSS] = dct[v];
  v2f bD[4];
#pragma unroll
  for (int kc = 0; kc < 4; ++kc) {
    bD[kc].x = buf[lB + kc * 4 * LDSS];          // pair LDSS apart -> ds_load_2addr_b32
    bD[kc].y = buf[lB + kc * 4 * LDSS + LDSS];
  }

  float acc[8];                                  // mid = x - low - high
#pragma unroll
  for (int v = 0; v < 8; ++v) acc[v] = cX[v];

#pragma unroll
  for (int m = 0; m < 2; ++m) {                  // m=0: low (band 0); m=1: high (band 2)
    v2f bm[4];
#pragma unroll
    for (int kc = 0; kc < 4; ++kc) bm[kc] = bD[kc] * (m ? mH[kc] : mL[kc]);

    // Stage 3: U = D2^T @ (DCT o mask)
    v8f u = {};
#pragma unroll
    for (int kc = 0; kc < 4; ++kc) u = wmma4(cQ[kc], bm[kc], u);

#pragma unroll
    for (int v = 0; v < 8; ++v) buf[lStore + v * LDSS] = u[v];
    v2f aU[4];
#pragma unroll
    for (int kc = 0; kc < 4; ++kc) {
      aU[kc].x = buf[lA + kc * 4];
      aU[kc].y = buf[lA + kc * 4 + 1];
    }

    // Stage 4: out_band = U @ D2
    v8f o = {};
#pragma unroll
    for (int kc = 0; kc < 4; ++kc) o = wmma4(aU[kc], cQ[kc], o);

    float* op = out + (m ? 2 : 0) * NTOT;
#pragma unroll
    for (int v = 0; v < 8; ++v) {
      __builtin_nontemporal_store(o[v], op + rowIdx + v * 512);
      acc[v] -= o[v];
    }
  }

  // mid band (output #1)
#pragma unroll
  for (int v = 0; v < 8; ++v)
    __builtin_nontemporal_store(acc[v], out + NTOT + rowIdx + v * 512);
}

extern "C" void kernel_launch(void* const* d_in, const int* in_sizes, int n_in,
                              void* d_out, int out_size, void* d_ws, size_t ws_size,
                              hipStream_t stream) {
  (void)in_sizes; (void)n_in; (void)ws_size; (void)out_size;
  const float* x = (const float*)d_in[0];
  float* out = (float*)d_out;
  v4f* ws = (v4f*)d_ws;                          // 4 KB of baked operand tables

  hipLaunchKernelGGL(freq_tabs_init, dim3(1), dim3(32), 0, stream, ws);

  // 16*3 planes of 512x512, 16x16 tiles -> 48*32*32 = 49152 tiles; 8 tiles per block
  const int tiles  = 49152;
  const int blocks = tiles / WPB;                // 6144
  hipLaunchKernelGGL(freq_decomp_wmma, dim3(blocks), dim3(256), 0, stream, x, out, ws);
}